// SetAbstraction_7335804142068
// MI455X (gfx1250) — compile-verified
//
#include <hip/hip_runtime.h>

// ---------------- problem constants ----------------
constexpr int cB = 8;
constexpr int cN = 16384;
constexpr int cS = 4096;
constexpr int cK = 32;
constexpr int cC = 64;
constexpr int cNP = cB * cS * cK;   // 1,048,576 "points" (columns of the MLP GEMM)

typedef __attribute__((ext_vector_type(2))) float v2f;
typedef __attribute__((ext_vector_type(8))) float v8f;

// ---------------- 1) gather new_xyz = xyz[:, :, perm] ----------------
__global__ __launch_bounds__(256) void k_newxyz(const float* __restrict__ xyz,
                                                const int* __restrict__ perm,
                                                float* __restrict__ out) {
    int i = blockIdx.x * blockDim.x + threadIdx.x;
    if (i >= cB * 3 * cS) return;
    int b = i / (3 * cS);
    int r = i % (3 * cS);
    int d = r / cS;
    int s = r % cS;
    out[i] = xyz[(size_t)b * 3 * cN + (size_t)d * cN + perm[s]];
}

// ---------------- 2) KNN: idx[s, 0..31] = 32 nearest support points ----------------
// One workgroup per query. Each of 256 threads owns a contiguous 64-point chunk,
// keeps its local best (with taken-mask) in LDS; 32 passes of a tree argmin with
// index tie-break (matches jax top_k ordering for -d2).
__global__ __launch_bounds__(256) void k_knn(const float* __restrict__ xyz,
                                             const float* __restrict__ nxyz,
                                             int* __restrict__ idxb) {
    const int s = blockIdx.x;
    const int t = threadIdx.x;
    const int n0 = t * 64;

    __shared__ float cd[256];
    __shared__ int   ci[256];
    __shared__ float rd[256];
    __shared__ int   ri[256];
    __shared__ int   ro[256];

    const float qx = nxyz[s];
    const float qy = nxyz[cS + s];
    const float qz = nxyz[2 * cS + s];
    const float* __restrict__ sx = xyz;
    const float* __restrict__ sy = xyz + cN;
    const float* __restrict__ sz = xyz + 2 * cN;

    unsigned long long taken = 0ull;

    float bd = 3.4e38f; int bi = -1;
    for (int j = 0; j < 64; ++j) {
        int n = n0 + j;
        float dx = sx[n] - qx, dy = sy[n] - qy, dz = sz[n] - qz;
        float d = dx * dx + dy * dy + dz * dz;
        if (d < bd) { bd = d; bi = n; }   // first hit wins -> smallest index on tie
    }
    cd[t] = bd; ci[t] = bi;

    for (int k = 0; k < cK; ++k) {
        __syncthreads();
        rd[t] = cd[t]; ri[t] = ci[t]; ro[t] = t;
        __syncthreads();
        for (int off = 128; off > 0; off >>= 1) {
            if (t < off) {
                float d2 = rd[t + off]; int i2 = ri[t + off];
                if (d2 < rd[t] || (d2 == rd[t] && i2 < ri[t])) {
                    rd[t] = d2; ri[t] = i2; ro[t] = ro[t + off];
                }
            }
            __syncthreads();
        }
        int win = ri[0];
        int owner = ro[0];
        if (t == 0) idxb[s * cK + k] = win;
        if (t == owner) {
            taken |= (1ull << (win - n0));
            float nb = 3.4e38f; int nbi = -1;
            for (int j = 0; j < 64; ++j) {
                if (taken & (1ull << j)) continue;
                int n = n0 + j;
                float dx = sx[n] - qx, dy = sy[n] - qy, dz = sz[n] - qz;
                float d = dx * dx + dy * dy + dz * dz;
                if (d < nb) { nb = d; nbi = n; }
            }
            cd[t] = nb; ci[t] = nbi;
        }
    }
}

// ---------------- 3) transpose feats (B,C,N) -> featsT (B,N,C) ----------------
__global__ __launch_bounds__(256) void k_transpose(const float* __restrict__ feats,
                                                   float* __restrict__ featsT) {
    __shared__ float tile[32][33];
    int b  = blockIdx.z;
    int n0 = blockIdx.x * 32;
    int c0 = blockIdx.y * 32;
    int tx = threadIdx.x;     // 0..31
    int ty = threadIdx.y;     // 0..7
    for (int i = 0; i < 32; i += 8)
        tile[ty + i][tx] = feats[(size_t)b * cC * cN + (size_t)(c0 + ty + i) * cN + n0 + tx];
    __syncthreads();
    for (int i = 0; i < 32; i += 8)
        featsT[(size_t)b * cN * cC + (size_t)(n0 + ty + i) * cC + c0 + tx] = tile[tx][ty + i];
}

// ---------------- 4) build X0: (NP, 68) point-major: [64 feats, 3 rel_xyz, 0 pad] ----------------
__global__ __launch_bounds__(256) void k_build_x0(const float* __restrict__ featsT,
                                                  const int* __restrict__ idxb,
                                                  const float* __restrict__ xyz,
                                                  const float* __restrict__ nxyz,
                                                  float* __restrict__ X0) {
    int p = blockIdx.x * 4 + (threadIdx.x >> 6);   // point index, 4 per block
    int c = threadIdx.x & 63;
    int b = p / (cS * cK);
    int r = p % (cS * cK);
    int s = r / cK;
    int n = idxb[r];   // r == s*K + k
    X0[(size_t)p * 68 + c] = featsT[(size_t)b * cN * cC + (size_t)n * cC + c];
    if (c < 4) {
        float v = 0.f;
        if (c < 3)
            v = xyz[(size_t)b * 3 * cN + (size_t)c * cN + n]
              - nxyz[(size_t)b * 3 * cS + (size_t)c * cS + s];
        X0[(size_t)p * 68 + 64 + c] = v;
    }
}

// ---------------- 5) GEMM: Y(point,Cout) = W(Cout,Cin) @ X(point,Cin)^T via fp32 WMMA ----------------
// Optional input normalization (BN+ReLU of the previous layer) folded into the B-operand load.
template <int CINP, int MTILES, bool HAS_BN>
__global__ __launch_bounds__(256) void k_gemm(const float* __restrict__ X,
                                              const float* __restrict__ W, int Cin,
                                              const float* __restrict__ nscale,
                                              const float* __restrict__ nshift,
                                              float* __restrict__ Y, int npoints) {
    constexpr int COUT = MTILES * 16;
    __shared__ float Ws[COUT * CINP];
    for (int i = threadIdx.x; i < COUT * CINP; i += blockDim.x) {
        int m = i / CINP, c = i % CINP;
        Ws[i] = (c < Cin) ? W[m * Cin + c] : 0.f;   // zero-pad channels
    }
    __syncthreads();

    const int lane  = threadIdx.x & 31;
    const int wave  = threadIdx.x >> 5;
    const int gwave = blockIdx.x * (blockDim.x >> 5) + wave;
    const int nwaves = gridDim.x * (blockDim.x >> 5);
    const int ntiles = npoints >> 4;

    const int col = lane & 15;         // N column within 16-wide tile
    const int kk  = (lane >> 4) << 1;  // lane half selects K {0,1} vs {2,3}

    for (int ct = gwave; ct < ntiles; ct += nwaves) {
        const int p = ct * 16 + col;
        const float* __restrict__ xp = X + (size_t)p * CINP;

        v8f acc[MTILES] = {};
#pragma unroll
        for (int kt = 0; kt < CINP / 4; ++kt) {
            const int c0 = kt * 4 + kk;
            float bx = xp[c0], by = xp[c0 + 1];
            if (HAS_BN) {
                bx = fmaxf(fmaf(bx, nscale[c0],     nshift[c0]),     0.f);
                by = fmaxf(fmaf(by, nscale[c0 + 1], nshift[c0 + 1]), 0.f);
            }
            v2f bv; bv.x = bx; bv.y = by;
#pragma unroll
            for (int mt = 0; mt < MTILES; ++mt) {
                const int m = mt * 16 + (lane & 15);
                v2f av; av.x = Ws[m * CINP + c0]; av.y = Ws[m * CINP + c0 + 1];
                acc[mt] = __builtin_amdgcn_wmma_f32_16x16x4_f32(
                    false, av, false, bv, (short)0, acc[mt], false, false);
            }
        }

        float* __restrict__ yp = Y + (size_t)p * COUT;
#pragma unroll
        for (int mt = 0; mt < MTILES; ++mt) {
            const int mbase = mt * 16 + ((lane >> 4) << 3);
#pragma unroll
            for (int r = 0; r < 8; ++r) yp[mbase + r] = acc[mt][r];
        }
    }
}

// ---------------- 6) per-channel sum / sumsq over raw conv output ----------------
__global__ __launch_bounds__(256) void k_stats(const float* __restrict__ Y, int npoints, int Cout,
                                               float* __restrict__ sum, float* __restrict__ sumsq) {
    int tid = blockIdx.x * blockDim.x + threadIdx.x;
    int total = gridDim.x * blockDim.x;
    int c = tid % Cout;
    int row0 = tid / Cout;
    int rstride = total / Cout;
    float s = 0.f, s2 = 0.f;
    for (int row = row0; row < npoints; row += rstride) {
        float v = Y[(size_t)row * Cout + c];
        s += v; s2 += v * v;
    }
    atomicAdd(&sum[c], s);
    atomicAdd(&sumsq[c], s2);
}

__global__ void k_bnfin(const float* __restrict__ sum, const float* __restrict__ sumsq,
                        const float* __restrict__ g, const float* __restrict__ bb,
                        float* __restrict__ nscale, float* __restrict__ nshift,
                        int Cout, float invcnt) {
    int c = threadIdx.x;
    if (c >= Cout) return;
    float mean = sum[c] * invcnt;
    float var  = sumsq[c] * invcnt - mean * mean;
    float rstd = rsqrtf(var + 1e-5f);
    float a = rstd * g[c];
    nscale[c] = a;
    nshift[c] = bb[c] - mean * a;
}

// ---------------- 7) final BN + ReLU + max over K ----------------
__global__ __launch_bounds__(128) void k_final(const float* __restrict__ Y3,
                                               const float* __restrict__ nscale,
                                               const float* __restrict__ nshift,
                                               float* __restrict__ outf) {
    int bs = blockIdx.x;           // b*S + s
    int c  = threadIdx.x;          // 0..127
    int b = bs / cS, s = bs % cS;
    const float* __restrict__ yp = Y3 + (size_t)bs * cK * 128;
    float a = nscale[c], sh = nshift[c];
    float m = -3.4e38f;
    for (int k = 0; k < cK; ++k) {
        float v = fmaxf(fmaf(yp[k * 128 + c], a, sh), 0.f);
        m = fmaxf(m, v);
    }
    outf[(size_t)b * 128 * cS + (size_t)c * cS + s] = m;
}

// ---------------- host launcher ----------------
extern "C" void kernel_launch(void* const* d_in, const int* in_sizes, int n_in,
                              void* d_out, int out_size, void* d_ws, size_t ws_size,
                              hipStream_t stream) {
    const float* xyz   = (const float*)d_in[0];
    const float* feats = (const float*)d_in[1];
    const int*   perm  = (const int*)d_in[2];
    const float* w0 = (const float*)d_in[3];
    const float* g0 = (const float*)d_in[4];
    const float* b0 = (const float*)d_in[5];
    const float* w1 = (const float*)d_in[6];
    const float* g1 = (const float*)d_in[7];
    const float* b1 = (const float*)d_in[8];
    const float* w2 = (const float*)d_in[9];
    const float* g2 = (const float*)d_in[10];
    const float* b2 = (const float*)d_in[11];

    float* out_newxyz = (float*)d_out;
    float* out_feats  = out_newxyz + (size_t)cB * 3 * cS;

    // workspace carve-up
    char* ws = (char*)d_ws;
    size_t off = 0;
    int* idxb = (int*)(ws + off);           off += (size_t)cS * cK * sizeof(int);
    float* sumb    = (float*)(ws + off);    off += 128 * sizeof(float);
    float* sumsqb  = (float*)(ws + off);    off += 128 * sizeof(float);
    float* nscaleb = (float*)(ws + off);    off += 128 * sizeof(float);
    float* nshiftb = (float*)(ws + off);    off += 128 * sizeof(float);
    off = (off + 255) & ~(size_t)255;
    float* featsT = (float*)(ws + off);     off += (size_t)cB * cN * cC * sizeof(float);
    float* X0     = (float*)(ws + off);     off += (size_t)cNP * 68 * sizeof(float); // reused as Y2
    float* Y1     = (float*)(ws + off);     off += (size_t)cNP * 64 * sizeof(float);
    float* Y3     = (float*)(ws + off);     off += (size_t)cNP * 128 * sizeof(float);
    float* Y2     = X0;  // X0 dead once Y1 is computed

    const float invcnt = 1.0f / (float)cNP;

    // 1) new_xyz (output #1, also consumed by KNN / rel_xyz)
    k_newxyz<<<(cB * 3 * cS + 255) / 256, 256, 0, stream>>>(xyz, perm, out_newxyz);
    // 2) KNN on batch-0 geometry
    k_knn<<<cS, 256, 0, stream>>>(xyz, out_newxyz, idxb);
    // 3) feats -> point-major
    k_transpose<<<dim3(cN / 32, cC / 32, cB), dim3(32, 8), 0, stream>>>(feats, featsT);
    // 4) X0 = concat(gathered feats, rel_xyz, pad)  (NP x 68)
    k_build_x0<<<cNP / 4, 256, 0, stream>>>(featsT, idxb, xyz, out_newxyz, X0);

    // layer 1: 67(->68) -> 64, raw input
    k_gemm<68, 4, false><<<1024, 256, 0, stream>>>(X0, w0, 67, nullptr, nullptr, Y1, cNP);
    hipMemsetAsync(sumb, 0, 2 * 128 * sizeof(float), stream);
    k_stats<<<256, 256, 0, stream>>>(Y1, cNP, 64, sumb, sumsqb);
    k_bnfin<<<1, 128, 0, stream>>>(sumb, sumsqb, g0, b0, nscaleb, nshiftb, 64, invcnt);

    // layer 2: 64 -> 64, BN1+ReLU folded into operand load
    k_gemm<64, 4, true><<<1024, 256, 0, stream>>>(Y1, w1, 64, nscaleb, nshiftb, Y2, cNP);
    hipMemsetAsync(sumb, 0, 2 * 128 * sizeof(float), stream);
    k_stats<<<256, 256, 0, stream>>>(Y2, cNP, 64, sumb, sumsqb);
    k_bnfin<<<1, 128, 0, stream>>>(sumb, sumsqb, g1, b1, nscaleb, nshiftb, 64, invcnt);

    // layer 3: 64 -> 128, BN2+ReLU folded into operand load
    k_gemm<64, 8, true><<<1024, 256, 0, stream>>>(Y2, w2, 64, nscaleb, nshiftb, Y3, cNP);
    hipMemsetAsync(sumb, 0, 2 * 128 * sizeof(float), stream);
    k_stats<<<256, 256, 0, stream>>>(Y3, cNP, 128, sumb, sumsqb);
    k_bnfin<<<1, 128, 0, stream>>>(sumb, sumsqb, g2, b2, nscaleb, nshiftb, 128, invcnt);

    // BN3 + ReLU + max over K -> new_feats (B,128,S)
    k_final<<<cB * cS, 128, 0, stream>>>(Y3, nscaleb, nshiftb, out_feats);
}